// opt_kde_11811160064316
// MI455X (gfx1250) — compile-verified
//
#include <hip/hip_runtime.h>
#include <hip/hip_bf16.h>
#include <math.h>

// ---------------------------------------------------------------------------
// Problem constants (fixed by the reference: E=4, L=10, S=1024, F=256, P=1024)
// ---------------------------------------------------------------------------
#define E_    4
#define L_    10
#define S_    1024
#define F_    256
#define P_    1024          // sample_size
#define PCH   8             // grid points per chunk (matches reference CHUNK)
#define NPC   (P_ / PCH)    // 128 p-chunks
#define NPAIR 6             // unique (e1<e2) pairs out of 4x4
#define LF    (L_ * F_)     // 2560

// bw = S^(-1/5) = 1024^(-0.2) = 0.25 exactly; c = -0.5/bw^2 = -8
static constexpr float C_COEF = -8.0f;
static constexpr float LOG2E  = 1.44269504088896340736f;
static constexpr float CC     = C_COEF * LOG2E;          // exp(c*t) == exp2(CC_base2 arg)
static constexpr float B2     = -2.0f * CC;              // +2*|CC|
static constexpr float DIVISOR = 0.62665706865775006f;   // sqrt(2*pi) * 0.25

typedef __attribute__((ext_vector_type(2))) float v2f;
typedef __attribute__((ext_vector_type(8))) float v8f;

// ----- float <-> order-preserving unsigned encoding for min/max atomics -----
__device__ __forceinline__ unsigned fenc(float f) {
    unsigned b = __float_as_uint(f);
    return (b & 0x80000000u) ? ~b : (b | 0x80000000u);
}
__device__ __forceinline__ float fdec(unsigned u) {
    unsigned b = (u & 0x80000000u) ? (u & 0x7FFFFFFFu) : ~u;
    return __uint_as_float(b);
}

// ---------------------------------------------------------------------------
// Kernel 0: initialize the encoded min/max cells in workspace
// ---------------------------------------------------------------------------
__global__ void k_init(unsigned* __restrict__ mm) {
    if (threadIdx.x == 0 && blockIdx.x == 0) {
        mm[0] = 0xFFFFFFFFu;  // encoded min accumulator
        mm[1] = 0x00000000u;  // encoded max accumulator
    }
}

// ---------------------------------------------------------------------------
// Kernel 1: m = matrix @ params via V_WMMA_F32_16X16X4_F32, fused global
//           min/max reduction. Rows = E*L*S = 40960, N = K = 256.
//           Block = 128 threads (4 waves), each wave a 16x16 tile.
// ---------------------------------------------------------------------------
__global__ __launch_bounds__(128) void k_gemm_minmax(
    const float* __restrict__ A,   // (40960, 256) row-major
    const float* __restrict__ B,   // (256, 256)   row-major
    float* __restrict__ M,         // (40960, 256) output
    unsigned* __restrict__ mm)
{
    const int lane = threadIdx.x & 31;
    const int wave = threadIdx.x >> 5;
    const int m0   = blockIdx.x * 64 + wave * 16;   // 640 blocks in x
    const int n0   = blockIdx.y * 16;               // 16 blocks in y
    const int half = lane >> 4;                     // 0: lanes 0-15, 1: 16-31
    const int l16  = lane & 15;
    const int row  = m0 + l16;                      // A row owned by this lane
    const int col  = n0 + l16;                      // B/C column owned by this lane
    const int kb   = half * 2;                      // K sub-offset (ISA layout)

    const float* __restrict__ Arow = A + (size_t)row * F_;

    v8f c = {0.f, 0.f, 0.f, 0.f, 0.f, 0.f, 0.f, 0.f};
    for (int k0 = 0; k0 < F_; k0 += 4) {
        // A 16x4 f32 fragment: VGPR0 = K {0 | 2}, VGPR1 = K {1 | 3}
        v2f a, b;
        a.x = Arow[k0 + kb];
        a.y = Arow[k0 + kb + 1];
        // B 4x16 f32 fragment, mirrored K striping across lane halves
        b.x = B[(size_t)(k0 + kb)     * F_ + col];
        b.y = B[(size_t)(k0 + kb + 1) * F_ + col];
        c = __builtin_amdgcn_wmma_f32_16x16x4_f32(
                /*neg_a=*/false, a, /*neg_b=*/false, b,
                /*c_mod=*/(short)0, c, /*reuse_a=*/false, /*reuse_b=*/false);
    }

    // C/D layout: VGPR r -> row m0+r (lanes 0-15) / m0+8+r (lanes 16-31)
    float lmin = c[0], lmax = c[0];
#pragma unroll
    for (int r = 0; r < 8; ++r) {
        const int mr = m0 + r + half * 8;
        M[(size_t)mr * F_ + col] = c[r];
        lmin = fminf(lmin, c[r]);
        lmax = fmaxf(lmax, c[r]);
    }

    // wave32 butterfly reduction, then one atomic pair per wave
#pragma unroll
    for (int off = 16; off >= 1; off >>= 1) {
        lmin = fminf(lmin, __shfl_xor(lmin, off, 32));
        lmax = fmaxf(lmax, __shfl_xor(lmax, off, 32));
    }
    if (lane == 0) {
        atomicMin(&mm[0], fenc(lmin));
        atomicMax(&mm[1], fenc(lmax));
    }
}

// ---------------------------------------------------------------------------
// Kernel 2: the KDE sweep. Block = 256 threads (one per f), one (l, p-chunk)
//           per block -> 1280 blocks. Quadratic expanded so each accumulation
//           is add + fma + v_exp_f32 + add (3 VALU + 1 TRANS), arg <= 0 so no
//           overflow hazard. Writes disjoint per-chunk partials (no atomics).
// ---------------------------------------------------------------------------
__global__ __launch_bounds__(256) void k_kde(
    const float* __restrict__ M,     // (E, L, S, F)
    const float* __restrict__ dlen,  // (E, L)
    const unsigned* __restrict__ mm,
    float* __restrict__ part)        // (NPC, NPAIR, L*F)
{
    const int f  = threadIdx.x;
    const int l  = blockIdx.x / NPC;
    const int pc = blockIdx.x % NPC;

    const float left  = fdec(mm[0]);
    const float right = fdec(mm[1]);
    const float step  = (right - left) / (float)(P_ - 1);   // linspace step

    float x[PCH], q[PCH];
#pragma unroll
    for (int p = 0; p < PCH; ++p) {
        x[p] = left + step * (float)(pc * PCH + p);
        q[p] = CC * x[p] * x[p];                   // hoisted out of the s-loop
    }

    float ks[E_][PCH];
#pragma unroll
    for (int e = 0; e < E_; ++e)
#pragma unroll
        for (int p = 0; p < PCH; ++p) ks[e][p] = 0.f;

    const size_t estride = (size_t)L_ * S_ * F_;
    const float* __restrict__ base = M + ((size_t)l * S_) * F_ + f;

    for (int s = 0; s < S_; ++s) {
        float mv[E_];
#pragma unroll
        for (int e = 0; e < E_; ++e)
            mv[e] = base[(size_t)e * estride + (size_t)s * F_];

        // speculative stream prefetch (global_prefetch_b8); safely dropped OOB
        __builtin_prefetch(base + (size_t)(s + 16) * F_, 0, 1);

#pragma unroll
        for (int e = 0; e < E_; ++e) {
            // CC*(mv-x)^2 = am + bm*x + q[p]
            const float am = CC * mv[e] * mv[e];
            const float bm = B2 * mv[e];
#pragma unroll
            for (int p = 0; p < PCH; ++p) {
                const float arg = fmaf(bm, x[p], am + q[p]);
                ks[e][p] += __builtin_amdgcn_exp2f(arg);   // v_exp_f32
            }
        }
    }

    // red = (ksum - (S - dl)*exp(c*x^2)) / dl ;  exp(c*x^2) = exp2(q[p])
    float red[E_][PCH];
#pragma unroll
    for (int e = 0; e < E_; ++e) {
        const float dlv = dlen[e * L_ + l];
        const float rdl = 1.0f / dlv;
        const float sml = (float)S_ - dlv;
#pragma unroll
        for (int p = 0; p < PCH; ++p) {
            const float zc = sml * __builtin_amdgcn_exp2f(q[p]);
            red[e][p] = (ks[e][p] - zc) * rdl;
        }
    }

    // 6 unique pairs; diagonal = 0 and matrix is symmetric under abs
    float pa[NPAIR] = {0.f, 0.f, 0.f, 0.f, 0.f, 0.f};
#pragma unroll
    for (int p = 0; p < PCH; ++p) {
        pa[0] += fabsf(red[0][p] - red[1][p]);
        pa[1] += fabsf(red[0][p] - red[2][p]);
        pa[2] += fabsf(red[0][p] - red[3][p]);
        pa[3] += fabsf(red[1][p] - red[2][p]);
        pa[4] += fabsf(red[1][p] - red[3][p]);
        pa[5] += fabsf(red[2][p] - red[3][p]);
    }

    const int j = l * F_ + f;
    float* __restrict__ dst = part + (size_t)pc * (NPAIR * LF);
#pragma unroll
    for (int k = 0; k < NPAIR; ++k)
        dst[k * LF + j] = pa[k];
}

// ---------------------------------------------------------------------------
// Kernel 3: deterministic reduction over p-chunks, scale, emit train||test
// ---------------------------------------------------------------------------
__global__ void k_final(
    const float* __restrict__ part,  // (NPC, NPAIR, L*F)
    const unsigned* __restrict__ mm,
    float* __restrict__ out)         // train (L*F) then test (L*F)
{
    const int j = blockIdx.x * blockDim.x + threadIdx.x;
    if (j >= LF) return;

    const float left  = fdec(mm[0]);
    const float right = fdec(mm[1]);
    const float delta = (right - left) / (float)P_;
    const float scale = delta * 0.5f / DIVISOR;

    float v[NPAIR] = {0.f, 0.f, 0.f, 0.f, 0.f, 0.f};
    for (int pc = 0; pc < NPC; ++pc) {
        const float* __restrict__ src = part + (size_t)pc * (NPAIR * LF);
#pragma unroll
        for (int k = 0; k < NPAIR; ++k)
            v[k] += src[k * LF + j];
    }
#pragma unroll
    for (int k = 0; k < NPAIR; ++k) v[k] *= scale;

    // test = max over all 16 (i,j): diagonal is 0, off-diagonals symmetric
    float test = 0.f;
#pragma unroll
    for (int k = 0; k < NPAIR; ++k) test = fmaxf(test, v[k]);
    // TRAIN_INDEX covers pairs among {0,2,3}: (0,2)=v[1], (0,3)=v[2], (2,3)=v[5]
    float train = fmaxf(fmaxf(v[1], v[2]), fmaxf(v[5], 0.f));

    out[j]      = train;   // train_results first (tuple order)
    out[LF + j] = test;    // test_results second
}

// ---------------------------------------------------------------------------
// Launch: init -> WMMA gemm+minmax -> KDE sweep -> finalize (all on stream)
//
// Workspace layout (floats):
//   [0..1]                      : encoded min/max (as unsigned)
//   [64   .. 64+1966080)        : per-chunk pair partials (128*6*2560 floats)
//   [1966144 .. +10485760)      : m = matrix @ params  (E*L*S, F)
// Total ~49.8 MB.
// ---------------------------------------------------------------------------
extern "C" void kernel_launch(void* const* d_in, const int* in_sizes, int n_in,
                              void* d_out, int out_size, void* d_ws, size_t ws_size,
                              hipStream_t stream)
{
    const float* matrix = (const float*)d_in[0];   // (4,10,1024,256) f32
    const float* dlen   = (const float*)d_in[1];   // (4,10) f32
    const float* params = (const float*)d_in[2];   // (256,256) f32
    // d_in[3] = sample_size (1024), fixed by the reference; hardcoded.

    unsigned* mm = (unsigned*)d_ws;
    float* wsf   = (float*)d_ws;
    float* part  = wsf + 64;                       // 128*6*2560 = 1,966,080 floats
    float* m     = wsf + 64 + (size_t)NPC * NPAIR * LF;  // 10,485,760 floats

    k_init<<<1, 32, 0, stream>>>(mm);

    dim3 ggemm((E_ * L_ * S_) / 64, F_ / 16);      // (640, 16)
    k_gemm_minmax<<<ggemm, 128, 0, stream>>>(matrix, params, m, mm);

    k_kde<<<L_ * NPC, 256, 0, stream>>>(m, dlen, mm, part);

    k_final<<<(LF + 255) / 256, 256, 0, stream>>>(part, mm, (float*)d_out);
}